// MIoU_48533130444893
// MI455X (gfx1250) — compile-verified
//
#include <hip/hip_runtime.h>
#include <hip/hip_bf16.h>

// MIoU (buggy-reference-faithful): result = popcount(presence(y_pred) over classes 1..20) / 21.
// Pure streaming reduction -> bandwidth bound (64 MiB once ~= 2.9us @ 23.3 TB/s). y_true dead.

#define NUM_CLASSES 21
// bits 1..20 of the presence mask
#define CLASS_MASK 0x001FFFFEu

// Native clang vector type: required for __builtin_nontemporal_load (HIP's int4 is a struct).
typedef int iv4 __attribute__((ext_vector_type(4)));

__device__ __forceinline__ unsigned mask4(iv4 v) {
    // Classes are in [0, 21) so (x & 31) == x; mask kept for safety on garbage input.
    return (1u << (v.x & 31)) | (1u << (v.y & 31)) |
           (1u << (v.z & 31)) | (1u << (v.w & 31));
}

__global__ void miou_init_kernel(unsigned* __restrict__ gmask) {
    if (threadIdx.x == 0 && blockIdx.x == 0) {
        *gmask = 0u;
    }
}

__global__ __launch_bounds__(256) void miou_hist_kernel(const iv4* __restrict__ yp4,
                                                        unsigned* __restrict__ gmask,
                                                        long long n4) {
    unsigned mask = 0u;
    const long long stride = (long long)gridDim.x * (long long)blockDim.x;
    long long i = (long long)blockIdx.x * (long long)blockDim.x + (long long)threadIdx.x;

    // Unrolled x4: four independent non-temporal global_load_b128 per body (clauseable),
    // each individually lane-coalesced. For 64*512*512 ints this is exactly one body.
    for (; i + 3 * stride < n4; i += 4 * stride) {
        iv4 a = __builtin_nontemporal_load(&yp4[i]);
        iv4 b = __builtin_nontemporal_load(&yp4[i + stride]);
        iv4 c = __builtin_nontemporal_load(&yp4[i + 2 * stride]);
        iv4 d = __builtin_nontemporal_load(&yp4[i + 3 * stride]);
        mask |= mask4(a);
        mask |= mask4(b);
        mask |= mask4(c);
        mask |= mask4(d);
    }
    // Remainder (0..3 strided chunks)
    for (; i < n4; i += stride) {
        mask |= mask4(__builtin_nontemporal_load(&yp4[i]));
    }

    // wave32 OR-reduction (5 butterfly steps)
    #pragma unroll
    for (int off = 16; off > 0; off >>= 1) {
        mask |= __shfl_xor(mask, off, 32);
    }

    __shared__ unsigned smask;
    if (threadIdx.x == 0) smask = 0u;
    __syncthreads();
    if ((threadIdx.x & 31u) == 0u) {
        atomicOr(&smask, mask);
    }
    __syncthreads();
    if (threadIdx.x == 0) {
        atomicOr(gmask, smask);   // one global atomic per workgroup (~4K total)
    }
}

__global__ void miou_finalize_kernel(const int* __restrict__ yp,
                                     long long tail_begin, long long n,
                                     const unsigned* __restrict__ gmask,
                                     float* __restrict__ out) {
    if (threadIdx.x == 0 && blockIdx.x == 0) {
        unsigned mask = *gmask;
        // scalar tail (n % 4 elements; zero for the 64*512*512 shape, kept for generality)
        for (long long i = tail_begin; i < n; ++i) {
            mask |= 1u << (yp[i] & 31);
        }
        unsigned present = __popc(mask & CLASS_MASK);
        out[0] = (float)present / (float)NUM_CLASSES;
    }
}

extern "C" void kernel_launch(void* const* d_in, const int* in_sizes, int n_in,
                              void* d_out, int out_size, void* d_ws, size_t ws_size,
                              hipStream_t stream) {
    const int* y_pred = (const int*)d_in[0];   // y_true (d_in[1]) is dead in the reference
    long long n = (long long)in_sizes[0];
    long long n4 = n >> 2;                     // # of int4 chunks

    unsigned* gmask = (unsigned*)d_ws;         // 4 bytes of scratch, re-zeroed every call
    float* out = (float*)d_out;

    miou_init_kernel<<<1, 1, 0, stream>>>(gmask);

    const int threads = 256;                   // 8 wave32s per block
    int blocks = 4096;                         // 32K waves; 4 x b128 loads per thread
    long long max_blocks = (n4 + threads - 1) / threads;
    if (max_blocks < 1) max_blocks = 1;
    if ((long long)blocks > max_blocks) blocks = (int)max_blocks;

    miou_hist_kernel<<<blocks, threads, 0, stream>>>((const iv4*)y_pred, gmask, n4);

    miou_finalize_kernel<<<1, 1, 0, stream>>>(y_pred, n4 << 2, n, gmask, out);
}